// MyEmbedding_88691074662625
// MI455X (gfx1250) — compile-verified
//
#include <hip/hip_runtime.h>

// Embedding gather: out[b,t,:] = table_b[text[b,t], :]
//   table_b = concat(weight[32000,D], figure_token_weight[2,D], vision_x[b,32,D])
// Pure bandwidth problem: ~134 MB out + ~134 MB in -> ~11 us @ 23.3 TB/s.
// WMMA (faithful one-hot GEMM) would be ~2.15 PFLOP fp32 + 2 GB of table
// streaming: ~1000x off the memory roofline, so no matrix ops by design.
//
// Data movement uses the CDNA5 async LDS engine:
//   global_load_async_to_lds_b128 (RT: table rows stay hot in 192MB L2)
//   -> s_wait_asynccnt 0 ->
//   global_store_async_from_lds_b128 (TH_STORE_NT: write-once output stream
//      must not evict the ~116 MB table working set)
// Row bytes never touch VGPRs; ASYNCcnt tracks completion; s_endpgm's
// implicit wait-idle drains the final stores.

#define VOCAB 32000
#define DDIM  4096
#define TVIS  32

typedef __attribute__((ext_vector_type(4))) float v4f;
typedef int v4i __attribute__((vector_size(16)));   // builtin param pointee type

// gfx12-class CPol immediate: TH in bits [2:0]; TH_STORE_NT == 1.
#define CPOL_RT       0
#define CPOL_STORE_NT 1

#if __has_builtin(__builtin_amdgcn_global_load_async_to_lds_b128) && \
    __has_builtin(__builtin_amdgcn_global_store_async_from_lds_b128)
#define USE_ASYNC_LDS 1
#else
#define USE_ASYNC_LDS 0
#endif

static __device__ __forceinline__ void wait_async_zero() {
#if __has_builtin(__builtin_amdgcn_s_wait_asynccnt)
    __builtin_amdgcn_s_wait_asynccnt(0);
#else
    asm volatile("s_wait_asynccnt 0x0" ::: "memory");
#endif
}

__global__ __launch_bounds__(256) void gather_embed_kernel(
    const float* __restrict__ vision_x,   // [B, TVIS, D]
    const float* __restrict__ weight,     // [VOCAB, D]
    const float* __restrict__ fig_w,      // [2, D]
    const int*   __restrict__ text,       // [B, T] flat
    float*       __restrict__ out,        // [B, T, D] flat
    unsigned     T)                       // tokens per batch row
{
    // 2D grid: x = token, y = batch. No integer division needed.
    const unsigned b   = blockIdx.y;
    const unsigned row = b * T + blockIdx.x;

    // Uniform address -> scalar load of the index (KMcnt path).
    const int idx = text[row];

    // Uniform (SALU) table select.
    const float* src;
    if (idx < VOCAB) {
        src = weight + (size_t)idx * DDIM;
    } else if (idx < VOCAB + 2) {
        src = fig_w + (size_t)(idx - VOCAB) * DDIM;
    } else {
        src = vision_x + ((size_t)b * TVIS + (size_t)(idx - (VOCAB + 2))) * DDIM;
    }
    float* dst = out + (size_t)row * DDIM;

    const v4f* __restrict__ s4 = (const v4f*)src;   // 1024 x 16B per row
    v4f*       __restrict__ d4 = (v4f*)dst;
    const unsigned t = threadIdx.x;                 // 0..255

#if USE_ASYNC_LDS
    __shared__ v4f buf[1024];                       // 16 KB row staging
    // Enqueue 4 async b128 loads per lane (ASYNCcnt += 4 per wave).
    #pragma unroll
    for (int c = 0; c < 4; ++c) {
        __builtin_amdgcn_global_load_async_to_lds_b128(
            (v4i*)(s4 + t + 256u * c), (v4i*)(buf + t + 256u * c),
            /*offset=*/0, /*cpol=*/CPOL_RT);
    }
    // Async loads complete in order; each lane stores exactly the LDS bytes
    // it loaded, so no workgroup barrier is required — only the counter wait.
    wait_async_zero();
    #pragma unroll
    for (int c = 0; c < 4; ++c) {
        __builtin_amdgcn_global_store_async_from_lds_b128(
            (v4i*)(d4 + t + 256u * c), (v4i*)(buf + t + 256u * c),
            /*offset=*/0, /*cpol=*/CPOL_STORE_NT);
    }
    // s_endpgm performs an implicit wait-idle, draining the async stores.
#else
    // Direct path: 4x b128 loads (single loadcnt wait), then stream the row
    // out with non-temporal b128 stores.
    v4f v0 = s4[t];
    v4f v1 = s4[t + 256];
    v4f v2 = s4[t + 512];
    v4f v3 = s4[t + 768];
    __builtin_nontemporal_store(v0, d4 + t);
    __builtin_nontemporal_store(v1, d4 + t + 256);
    __builtin_nontemporal_store(v2, d4 + t + 512);
    __builtin_nontemporal_store(v3, d4 + t + 768);
#endif
}

extern "C" void kernel_launch(void* const* d_in, const int* in_sizes, int n_in,
                              void* d_out, int out_size, void* d_ws, size_t ws_size,
                              hipStream_t stream)
{
    // setup_inputs() order: vision_x, weight, figure_token_weight, text_input
    const float* vision = (const float*)d_in[0];
    const float* weight = (const float*)d_in[1];
    const float* fig    = (const float*)d_in[2];
    const int*   text   = (const int*)d_in[3];
    float*       out    = (float*)d_out;

    const unsigned B = (unsigned)(in_sizes[0] / (TVIS * DDIM));  // 4
    const unsigned T = (unsigned)(in_sizes[3] / (int)B);         // 2048

    gather_embed_kernel<<<dim3(T, B), dim3(256), 0, stream>>>(
        vision, weight, fig, text, out, T);
}